// DETRScene_44624710205890
// MI455X (gfx1250) — compile-verified
//
#include <hip/hip_runtime.h>

typedef float v2f __attribute__((ext_vector_type(2)));
typedef float v8f __attribute__((ext_vector_type(8)));

#define L_  6
#define B_  256
#define Q1_ 101
#define D_  256
#define M_  64
#define R_  64
#define P_  51
#define K2_ (2 * D_)   // 512 : concat(sub, obj)

// ---------------------------------------------------------------------------
// Prep: transpose + pad W_pred (51 x 512, row-major) into a WMMA-B-friendly
// layout. For V_WMMA_F32_16X16X4_F32 the B matrix (4x16, K x N) is striped:
// lanes 0-15 -> N, lanes 16-31 -> N with K+2; each lane holds two consecutive
// K values in one VGPR pair. So we store K-pairs contiguously per column:
//   Wt2[(k>>1)*128 + n*2 + (k&1)] = (n < 51) ? W[n*512 + k] : 0
// giving each lane a single coalesced v2f load per (ktile, ntile).
// ---------------------------------------------------------------------------
__global__ void prep_w(const float* __restrict__ W, float* __restrict__ Wt2) {
    int idx = blockIdx.x * blockDim.x + threadIdx.x;   // over 512*64
    if (idx >= K2_ * 64) return;
    int k = idx >> 6;          // 0..511
    int n = idx & 63;          // 0..63 (padded N)
    float v = (n < P_) ? W[n * K2_ + k] : 0.0f;
    Wt2[(k >> 1) * 128 + n * 2 + (k & 1)] = v;
}

// ---------------------------------------------------------------------------
// Main kernel: one workgroup per (l,b). 4 waves; wave w owns output rows
// [16w, 16w+16) and all four 16-wide N tiles (64 cols, 51 valid).
// ---------------------------------------------------------------------------
__global__ __launch_bounds__(128) void detr_pair_gemm(
    const float* __restrict__ hs,          // (L,B,Q1,D)
    const int*   __restrict__ src_indices, // (L,B,M)
    const int*   __restrict__ tgt_perm,    // (L,B,M)
    const int*   __restrict__ rel,         // (B,R,2)
    const float* __restrict__ bias,        // (P)
    const float* __restrict__ Wt2,         // (512/2, 64, 2) padded/transposed
    float*       __restrict__ out)         // (L,B,R,P)
{
    __shared__ int lookup[M_];
    __shared__ int sq[R_];
    __shared__ int oq[R_];

    const int lb = blockIdx.x;             // 0 .. L*B-1
    const int b  = lb % B_;
    const int t  = threadIdx.x;

    // ---- index chase (divergent section, before any WMMA) ----
    const int* tp = tgt_perm + (size_t)lb * M_;
    if (t < M_) lookup[tp[t]] = t;         // inverse permutation
    __syncthreads();
    if (t < R_) {
        const int* si = src_indices + (size_t)lb * M_;
        int st = rel[((size_t)b * R_ + t) * 2 + 0];
        int ot = rel[((size_t)b * R_ + t) * 2 + 1];
        sq[t] = si[lookup[st]];
        oq[t] = si[lookup[ot]];
    }
    __syncthreads();

    const int lane = t & 31;
    const int wid  = t >> 5;
    const int m0   = wid * 16;
    const int mloc = lane & 15;
    const int koff = (lane >> 4) << 1;     // lanes 16-31 handle K+2 (ISA A/B layout)

    const float* hsb    = hs + (size_t)lb * Q1_ * D_;
    const float* subRow = hsb + (size_t)sq[m0 + mloc] * D_;
    const float* objRow = hsb + (size_t)oq[m0 + mloc] * D_;

    v8f acc[4];
    acc[0] = v8f{}; acc[1] = v8f{}; acc[2] = v8f{}; acc[3] = v8f{};

    // K in [0,256): sub half of pair;  K in [256,512): obj half.
    #pragma unroll 1
    for (int half = 0; half < 2; ++half) {
        const float* rowp  = half ? objRow : subRow;
        const float* wbase = Wt2 + (size_t)half * (D_ / 2) * 128;
        #pragma unroll 4
        for (int k = 0; k < D_; k += 4) {
            // A fragment: pair[m][k+koff], pair[m][k+koff+1]  (contiguous b64)
            v2f a = *(const v2f*)(rowp + k + koff);
            // B fragments: column n = nt*16 + (lane&15), K pair at k+koff
            const float* wrow = wbase + (size_t)((k + koff) >> 1) * 128 + mloc * 2;
            v2f b0 = *(const v2f*)(wrow + 0);
            v2f b1 = *(const v2f*)(wrow + 32);
            v2f b2 = *(const v2f*)(wrow + 64);
            v2f b3 = *(const v2f*)(wrow + 96);
            acc[0] = __builtin_amdgcn_wmma_f32_16x16x4_f32(false, a, false, b0,
                                                           (short)0, acc[0], false, false);
            acc[1] = __builtin_amdgcn_wmma_f32_16x16x4_f32(false, a, false, b1,
                                                           (short)0, acc[1], false, false);
            acc[2] = __builtin_amdgcn_wmma_f32_16x16x4_f32(false, a, false, b2,
                                                           (short)0, acc[2], false, false);
            acc[3] = __builtin_amdgcn_wmma_f32_16x16x4_f32(false, a, false, b3,
                                                           (short)0, acc[3], false, false);
        }
    }

    // ---- epilogue: D layout = VGPR j -> row m0 + j + (lane>=16 ? 8 : 0),
    //      col = nt*16 + (lane&15). Add bias, mask padded cols. ----
    const int prow = (lane >> 4) << 3;     // 0 or 8
    const int pcol = lane & 15;
    #pragma unroll
    for (int nt = 0; nt < 4; ++nt) {
        int p = nt * 16 + pcol;
        if (p < P_) {
            float bs = bias[p];
            #pragma unroll
            for (int j = 0; j < 8; ++j) {
                int r = m0 + prow + j;
                out[((size_t)lb * R_ + r) * P_ + p] = acc[nt][j] + bs;
            }
        }
    }
}

// ---------------------------------------------------------------------------
extern "C" void kernel_launch(void* const* d_in, const int* in_sizes, int n_in,
                              void* d_out, int out_size, void* d_ws, size_t ws_size,
                              hipStream_t stream) {
    const float* hs          = (const float*)d_in[0];
    const int*   src_indices = (const int*)  d_in[1];
    const int*   tgt_perm    = (const int*)  d_in[2];
    const int*   rel         = (const int*)  d_in[3];
    const float* W_pred      = (const float*)d_in[4];
    const float* b_pred      = (const float*)d_in[5];
    float*       out         = (float*)d_out;
    float*       Wt2         = (float*)d_ws;   // 512*64 floats = 128 KB

    prep_w<<<(K2_ * 64 + 255) / 256, 256, 0, stream>>>(W_pred, Wt2);
    detr_pair_gemm<<<L_ * B_, 128, 0, stream>>>(hs, src_indices, tgt_perm, rel,
                                                b_pred, Wt2, out);
}